// AntecedentScore_65463891526396
// MI455X (gfx1250) — compile-verified
//
#include <hip/hip_runtime.h>
#include <hip/hip_bf16.h>

typedef __attribute__((ext_vector_type(2))) float v2f;
typedef __attribute__((ext_vector_type(8))) float v8f;

#define KSPAN 1200
#define CDIM  150
#define DDIM  500
#define FDIM  20
#define HDIM  150
#define PE    1520          // 3*D + F
#define NPAD  160           // H padded to 10 WMMA N-tiles
#define NT    10
#define MTOT  (KSPAN*CDIM)  // 180000 = 11250 M-tiles of 16
#define NMTILE 11250
#define SLAB  80            // k-values per LDS slab for GEMM1 (1520 = 19*80)
#define SLAB_STEPS 20       // 80/4
#define NSLAB 19
#define WAVES 4             // M-tiles per workgroup

// ---------------------------------------------------------------------------
// Kernel 1: build antecedent_emb and pair_emb (pure bandwidth, float4 moves)
// ---------------------------------------------------------------------------
__global__ __launch_bounds__(128) void pair_build_kernel(
    const float* __restrict__ emb,       // [1200,500]
    const int*   __restrict__ ante,      // [1200,150] flat
    const float* __restrict__ dist_emb,  // [10,20]
    float* __restrict__ ant_out,         // [MTOT,500]
    float* __restrict__ pair_out)        // [MTOT,1520]
{
    const long long p = blockIdx.x;          // pair index
    const int tid = threadIdx.x;
    const int i = (int)(p / CDIM);
    const int j = ante[p];

    // bucket_distance(i - j)
    int d = i - j;
    int b;
    if (d <= 4) { b = d < 0 ? 0 : d; }
    else        { b = 34 - __clz(d); if (b > 9) b = 9; }

    const float4* T  = (const float4*)(emb + (long long)i * DDIM);
    const float4* A  = (const float4*)(emb + (long long)j * DDIM);
    float4* PR = (float4*)(pair_out + p * (long long)PE);
    float4* AR = (float4*)(ant_out  + p * (long long)DDIM);

    if (tid < 125) {                         // 125 float4 = 500 floats
        float4 t = T[tid];
        float4 a = A[tid];
        float4 s; s.x = t.x*a.x; s.y = t.y*a.y; s.z = t.z*a.z; s.w = t.w*a.w;
        PR[tid]       = t;                   // target_rep
        PR[125 + tid] = a;                   // antecedent
        PR[250 + tid] = s;                   // similarity
        AR[tid]       = a;                   // antecedent_emb output
    }
    if (tid < 5) {                           // 5 float4 = 20 feature floats
        PR[375 + tid] = ((const float4*)(dist_emb + b * FDIM))[tid];
    }
}

// ---------------------------------------------------------------------------
// Kernel 2: MLP scores via V_WMMA_F32_16X16X4_F32.
// 4 waves / block, 1 M-tile (16 pair rows) per wave, N = 160 (10 tiles).
// LDS B-operands pre-swizzled to fragment order:
//   idx = ((kstep*NT + ntile)*64 + lane*2 + d),  value = W[k + khalf + d][n]
// so a lane's B fragment is one aligned 8-byte LDS read (pairs of tiles merge
// into ds_load_2addr_b64).
// ---------------------------------------------------------------------------
__global__ __launch_bounds__(32*WAVES) void scores_mlp_kernel(
    const float* __restrict__ pair_emb,  // [MTOT,1520] (region of d_out)
    const float* __restrict__ W_in,      // [1520,150]
    const float* __restrict__ b_in,      // [150]
    const float* __restrict__ W_h,       // [150,150]
    const float* __restrict__ b_h,       // [150]
    const float* __restrict__ W_out,     // [150]
    const float* __restrict__ b_out,     // [1]
    const float* __restrict__ ms,        // [1200] mention scores
    const int*   __restrict__ ante,      // [1200,150] flat
    float* __restrict__ scores)          // [MTOT]
{
    __shared__ float lds_w[40 * NT * 64];          // 100 KB (GEMM1 slab / all W_h)
    __shared__ float lds_h1[WAVES * 16 * NPAD];    // 40 KB

    const int tid   = threadIdx.x;
    const int nthr  = 32 * WAVES;
    const int wave  = tid >> 5;
    const int lane  = tid & 31;
    const int l15   = lane & 15;
    const int khalf = (lane >> 4) << 1;            // 0 or 2
    const int mt    = blockIdx.x * WAVES + wave;   // may exceed NMTILE-1 on tail
    const int mtc   = mt < NMTILE ? mt : NMTILE - 1;   // clamped for loads
    const long long mb = (long long)mtc * 16;

    const v8f vzero = {0.f,0.f,0.f,0.f,0.f,0.f,0.f,0.f};
    v8f acc[NT];
    #pragma unroll
    for (int t = 0; t < NT; ++t) acc[t] = vzero;

    // A-row pointer: lane holds M = l15, K-pair = khalf + {0,1}
    const float* arow = pair_emb + (mb + l15) * (long long)PE + khalf;

    // ---------------- GEMM1: h1 = pair_emb @ W_in ----------------
    for (int slab = 0; slab < NSLAB; ++slab) {
        const int k0 = slab * SLAB;
        // stage: q enumerates (kc, nt, li); one b64 LDS store of {W[kk][n], W[kk+1][n]}
        for (int q = tid; q < SLAB_STEPS * NT * 32; q += nthr) {
            int kc  = q / (NT * 32);
            int rem = q % (NT * 32);
            int nt  = rem >> 5;
            int li  = rem & 31;
            int kk  = k0 + kc * 4 + ((li >> 4) << 1);
            int n   = nt * 16 + (li & 15);
            v2f w;
            w.x = (n < HDIM) ? W_in[kk * HDIM + n]       : 0.f;
            w.y = (n < HDIM) ? W_in[(kk + 1) * HDIM + n] : 0.f;
            ((v2f*)lds_w)[q] = w;
        }
        __syncthreads();
        for (int s = 0; s < SLAB_STEPS; ++s) {
            v2f a = *(const v2f*)(arow + k0 + s * 4);
            #pragma unroll
            for (int t = 0; t < NT; ++t) {
                v2f bfrag = *(const v2f*)&lds_w[(s * NT + t) * 64 + lane * 2];
                acc[t] = __builtin_amdgcn_wmma_f32_16x16x4_f32(
                    false, a, false, bfrag, (short)0, acc[t], false, false);
            }
        }
        __syncthreads();
    }

    // ---------------- bias + ReLU, h1 -> LDS ----------------
    float* h1 = &lds_h1[wave * 16 * NPAD];
    #pragma unroll
    for (int t = 0; t < NT; ++t) {
        int n = t * 16 + l15;
        float bv = (n < HDIM) ? b_in[n] : 0.f;
        #pragma unroll
        for (int r = 0; r < 8; ++r) {
            float v = acc[t][r] + bv;
            v = v > 0.f ? v : 0.f;
            int m = r + ((lane >> 4) << 3);
            h1[m * NPAD + n] = v;
        }
    }
    // stage all of W_h (padded 160x160) in fragment order
    for (int q = tid; q < 40 * NT * 32; q += nthr) {
        int kc  = q / (NT * 32);
        int rem = q % (NT * 32);
        int nt  = rem >> 5;
        int li  = rem & 31;
        int kk  = kc * 4 + ((li >> 4) << 1);
        int n   = nt * 16 + (li & 15);
        v2f w;
        w.x = (kk     < HDIM && n < HDIM) ? W_h[kk * HDIM + n]       : 0.f;
        w.y = (kk + 1 < HDIM && n < HDIM) ? W_h[(kk + 1) * HDIM + n] : 0.f;
        ((v2f*)lds_w)[q] = w;
    }
    __syncthreads();

    // ---------------- GEMM2: h2 = h1 @ W_h ----------------
    v8f acc2[NT];
    #pragma unroll
    for (int t = 0; t < NT; ++t) acc2[t] = vzero;
    for (int s = 0; s < 40; ++s) {
        v2f a = *(const v2f*)&h1[l15 * NPAD + s * 4 + khalf];
        #pragma unroll
        for (int t = 0; t < NT; ++t) {
            v2f bfrag = *(const v2f*)&lds_w[(s * NT + t) * 64 + lane * 2];
            acc2[t] = __builtin_amdgcn_wmma_f32_16x16x4_f32(
                false, a, false, bfrag, (short)0, acc2[t], false, false);
        }
    }

    // ---------------- bias + ReLU + dot(W_out) + reduce ----------------
    float part[8];
    #pragma unroll
    for (int r = 0; r < 8; ++r) part[r] = 0.f;
    #pragma unroll
    for (int t = 0; t < NT; ++t) {
        int n = t * 16 + l15;
        float bv = (n < HDIM) ? b_h[n]   : 0.f;
        float wv = (n < HDIM) ? W_out[n] : 0.f;
        #pragma unroll
        for (int r = 0; r < 8; ++r) {
            float v = acc2[t][r] + bv;
            v = v > 0.f ? v : 0.f;
            part[r] += v * wv;
        }
    }
    const float bo = b_out[0];
    #pragma unroll
    for (int r = 0; r < 8; ++r) {
        float v = part[r];
        v += __shfl_xor(v, 1, 32);
        v += __shfl_xor(v, 2, 32);
        v += __shfl_xor(v, 4, 32);
        v += __shfl_xor(v, 8, 32);
        if (l15 == 0 && mt < NMTILE) {
            int m = r + ((lane >> 4) << 3);
            long long p = (long long)mt * 16 + m;
            int i = (int)(p / CDIM);
            int j = ante[p];
            scores[p] = v + bo + ms[i] + ms[j];
        }
    }
}

// ---------------------------------------------------------------------------
extern "C" void kernel_launch(void* const* d_in, const int* in_sizes, int n_in,
                              void* d_out, int out_size, void* d_ws, size_t ws_size,
                              hipStream_t stream) {
    const float* emb  = (const float*)d_in[0];  // top_span_emb [1200,500]
    const float* msc  = (const float*)d_in[1];  // mention scores [1200]
    const int*   ante = (const int*)  d_in[2];  // antecedents [1200,150]
    const float* de   = (const float*)d_in[3];  // dist_emb [10,20]
    const float* Win  = (const float*)d_in[4];  // [1520,150]
    const float* bin  = (const float*)d_in[5];  // [150]
    const float* Wh   = (const float*)d_in[6];  // [150,150]
    const float* bh   = (const float*)d_in[7];  // [150]
    const float* Wo   = (const float*)d_in[8];  // [150,1]
    const float* bo   = (const float*)d_in[9];  // [1]

    float* out      = (float*)d_out;
    float* scores   = out;                                  // 180000
    float* ant_out  = out + (long long)MTOT;                // 90,000,000
    float* pair_out = ant_out + (long long)MTOT * DDIM;     // 273,600,000

    pair_build_kernel<<<MTOT, 128, 0, stream>>>(emb, ante, de, ant_out, pair_out);

    const int nblocks = (NMTILE + WAVES - 1) / WAVES;       // 2813 (tail-guarded)
    scores_mlp_kernel<<<nblocks, 32 * WAVES, 0, stream>>>(
        pair_out, Win, bin, Wh, bh, Wo, bo, msc, ante, scores);
}